// RelationalGraphLayer_32581621907910
// MI455X (gfx1250) — compile-verified
//
#include <hip/hip_runtime.h>

// Sizes fixed by the reference
#define N_NODES 50000
#define N_EDGES 200000
#define D_DIM   128
#define R_REL   4
#define H_DIM   64
#define NIN_W   640            // (R+1)*D

#define EW      4              // waves per block in MLP kernels (wave32)
#define XS_LD   132            // 16x128 tile, padded stride -> conflict-free (bank = (4m+c)%64)
#define HS_LD   68             // 16x64 tile, padded stride

typedef __attribute__((ext_vector_type(2))) float v2f;
typedef __attribute__((ext_vector_type(8))) float v8f;
typedef __attribute__((ext_vector_type(4))) int   v4i;

static __device__ __forceinline__ int imin(int a, int b) { return a < b ? a : b; }

// CDNA5 fp32 WMMA: D(16x16 f32) = A(16x4 f32) * B(4x16 f32) + C
// A frag (per lane): .x -> K=2*half, .y -> K=2*half+1, M = lane%16
// B frag (per lane): .x -> K=2*half, .y -> K=2*half+1, N = lane%16
// C/D frag: VGPR v -> M = v + 8*half, N = lane%16
static __device__ __forceinline__ v8f wmma4(v2f a, v2f b, v8f c) {
  return __builtin_amdgcn_wmma_f32_16x16x4_f32(false, a, false, b, (short)0, c, false, false);
}

// ---- gfx1250 async memory->LDS staging (ASYNCcnt path), guarded so compile
// ---- cannot break if the builtin is absent; fallback = load+store via VGPRs.
#if defined(__AMDGCN__) && defined(__gfx1250__) && \
    __has_builtin(__builtin_amdgcn_global_load_async_to_lds_b128) && \
    __has_builtin(__builtin_amdgcn_s_wait_asynccnt)
#define USE_ASYNC_LDS 1
typedef __attribute__((address_space(1))) v4i g_v4i;   // global int4
typedef __attribute__((address_space(3))) v4i l_v4i;   // LDS int4
#else
#define USE_ASYNC_LDS 0
#endif

// copy 64 consecutive floats (one half X-row) global -> LDS (16B granules)
static __device__ __forceinline__ void stage_row64(const float* __restrict__ grow,
                                                   float* __restrict__ lrow) {
#if USE_ASYNC_LDS
#pragma unroll
  for (int i = 0; i < 16; ++i) {
    __builtin_amdgcn_global_load_async_to_lds_b128(
        (g_v4i*)(grow + 4 * i), (l_v4i*)(lrow + 4 * i), 0, 0);
  }
#else
#pragma unroll
  for (int i = 0; i < 16; ++i) {
    float4 v = ((const float4*)grow)[i];
    lrow[4 * i + 0] = v.x; lrow[4 * i + 1] = v.y;
    lrow[4 * i + 2] = v.z; lrow[4 * i + 3] = v.w;
  }
#endif
}

static __device__ __forceinline__ void async_fence() {
#if USE_ASYNC_LDS
  __builtin_amdgcn_s_wait_asynccnt(0);
#endif
}

// ---------------- small prep kernels ----------------

__global__ void zero_meta_kernel(int* meta) {
  if (blockIdx.x == 0 && threadIdx.x < 16) meta[threadIdx.x] = 0;
}

// one block per node: node_in[n,0:128]=relu(x[n]); node_in[n,128:640]=0
__global__ void init_nodein_kernel(const float* __restrict__ nodef,
                                   float* __restrict__ node_in) {
  int n = blockIdx.x;
  int c = threadIdx.x;
  node_in[(size_t)n * NIN_W + c] =
      (c < D_DIM) ? fmaxf(nodef[(size_t)n * D_DIM + c], 0.0f) : 0.0f;
}

__global__ void count_kernel(const int* __restrict__ etype, int* meta) {
  int e = blockIdx.x * blockDim.x + threadIdx.x;
  if (e < N_EDGES) atomicAdd(&meta[etype[e]], 1);
}

__global__ void scan_kernel(int* meta) {
  if (blockIdx.x == 0 && threadIdx.x == 0) {
    meta[4] = 0;
    for (int r = 0; r < R_REL; ++r) meta[5 + r] = meta[4 + r] + meta[r];
  }
}

__global__ void scatter_kernel(const int* __restrict__ etype, int* meta,
                               int* __restrict__ bucket) {
  int e = blockIdx.x * blockDim.x + threadIdx.x;
  if (e < N_EDGES) {
    int r = etype[e];
    int pos = atomicAdd(&meta[9 + r], 1);
    bucket[meta[4 + r] + pos] = e;
  }
}

// ---------------- edge message MLP (WMMA) ----------------
// One wave handles 16 edges of a single relation.
__global__ __launch_bounds__(EW * 32) void edge_mlp_kernel(
    const float* __restrict__ nodef,
    const int* __restrict__ esrc, const int* __restrict__ edst,
    const float* __restrict__ Wr1, const float* __restrict__ br1,
    const float* __restrict__ Wr2, const float* __restrict__ br2,
    const float* __restrict__ Wr3, const float* __restrict__ br3,
    const int* __restrict__ bucket, const int* __restrict__ meta,
    float* __restrict__ node_in) {
  __shared__ __align__(16) float Xs[EW][16 * XS_LD];
  __shared__ __align__(16) float Hs[EW][16 * HS_LD];
  __shared__ int Ds[EW][16];

  const int tid  = threadIdx.x;
  const int w    = tid >> 5;
  const int lane = tid & 31;
  const int half = lane >> 4;
  const int ln   = lane & 15;

  // map wave -> (relation, group-within-relation) from device-side counts
  const int g = blockIdx.x * EW + w;
  int rel = -1, lg = 0, off = 0, cnt = 0;
  {
    int acc = 0;
#pragma unroll
    for (int r = 0; r < R_REL; ++r) {
      int c  = meta[r];
      int ng = (c + 15) >> 4;
      if (rel < 0 && g < acc + ng) { rel = r; lg = g - acc; off = meta[4 + r]; cnt = c; }
      acc += ng;
    }
  }
  const bool active = (rel >= 0);
  const int  relc   = active ? rel : 0;
  const int  gcount = active ? imin(16, cnt - lg * 16) : 0;

  float* xs = &Xs[w][0];
  float* hs = &Hs[w][0];

  // stage gathered X rows [16 x 128] + destinations
  if (active) {
    if (lane < 16) {
      int t  = lane;
      int tt = imin(t, gcount - 1);
      int e  = bucket[off + lg * 16 + tt];
      Ds[w][t] = (t < gcount) ? edst[e] : -1;
    }
    int row = lane >> 1;
    int rr  = imin(row, gcount - 1);
    int e2  = bucket[off + lg * 16 + rr];
    int src = esrc[e2];
    stage_row64(nodef + (size_t)src * D_DIM + (lane & 1) * 64,
                xs + row * XS_LD + (lane & 1) * 64);
  }
  async_fence();
  __syncthreads();

  // ---- layer 1: [16x128] @ [128x64] ----
  const float* W1 = Wr1 + (size_t)relc * D_DIM * H_DIM;
  v8f c1[4];
#pragma unroll
  for (int j = 0; j < 4; ++j)
#pragma unroll
    for (int i = 0; i < 8; ++i) c1[j][i] = 0.0f;

  for (int kk = 0; kk < 32; ++kk) {
    int krow = kk * 4 + half * 2;
    v2f a = *(const v2f*)(xs + ln * XS_LD + krow);
#pragma unroll
    for (int j = 0; j < 4; ++j) {
      v2f b;
      b.x = W1[(size_t)(krow + 0) * H_DIM + j * 16 + ln];
      b.y = W1[(size_t)(krow + 1) * H_DIM + j * 16 + ln];
      c1[j] = wmma4(a, b, c1[j]);
    }
  }
  {
    const float* b1 = br1 + relc * H_DIM;
#pragma unroll
    for (int j = 0; j < 4; ++j) {
      float bias = b1[j * 16 + ln];
#pragma unroll
      for (int v = 0; v < 8; ++v)
        hs[(v + 8 * half) * HS_LD + j * 16 + ln] = fmaxf(c1[j][v] + bias, 0.0f);
    }
  }
  __syncthreads();

  // ---- layer 2: [16x64] @ [64x64] ----
  const float* W2 = Wr2 + (size_t)relc * H_DIM * H_DIM;
  v8f c2[4];
#pragma unroll
  for (int j = 0; j < 4; ++j)
#pragma unroll
    for (int i = 0; i < 8; ++i) c2[j][i] = 0.0f;

  for (int kk = 0; kk < 16; ++kk) {
    int krow = kk * 4 + half * 2;
    v2f a = *(const v2f*)(hs + ln * HS_LD + krow);
#pragma unroll
    for (int j = 0; j < 4; ++j) {
      v2f b;
      b.x = W2[(size_t)(krow + 0) * H_DIM + j * 16 + ln];
      b.y = W2[(size_t)(krow + 1) * H_DIM + j * 16 + ln];
      c2[j] = wmma4(a, b, c2[j]);
    }
  }
  __syncthreads();
  {
    const float* b2 = br2 + relc * H_DIM;
#pragma unroll
    for (int j = 0; j < 4; ++j) {
      float bias = b2[j * 16 + ln];
#pragma unroll
      for (int v = 0; v < 8; ++v)
        xs[(v + 8 * half) * XS_LD + j * 16 + ln] = fmaxf(c2[j][v] + bias, 0.0f);
    }
  }
  __syncthreads();

  // ---- layer 3: [16x64] @ [64x128] ----
  const float* W3 = Wr3 + (size_t)relc * H_DIM * D_DIM;
  v8f c3[8];
#pragma unroll
  for (int j = 0; j < 8; ++j)
#pragma unroll
    for (int i = 0; i < 8; ++i) c3[j][i] = 0.0f;

  for (int kk = 0; kk < 16; ++kk) {
    int krow = kk * 4 + half * 2;
    v2f a = *(const v2f*)(xs + ln * XS_LD + krow);
#pragma unroll
    for (int j = 0; j < 8; ++j) {
      v2f b;
      b.x = W3[(size_t)(krow + 0) * D_DIM + j * 16 + ln];
      b.y = W3[(size_t)(krow + 1) * D_DIM + j * 16 + ln];
      c3[j] = wmma4(a, b, c3[j]);
    }
  }

  // relu(msg) scatter-add into node_in[dst, 128 + r*128 + col]
  if (active) {
    const float* b3 = br3 + relc * D_DIM;
    const int colbase = D_DIM + relc * D_DIM;
#pragma unroll
    for (int j = 0; j < 8; ++j) {
      float bias = b3[j * 16 + ln];
#pragma unroll
      for (int v = 0; v < 8; ++v) {
        int m  = v + 8 * half;
        int dn = Ds[w][m];
        if (dn >= 0) {
          float val = fmaxf(c3[j][v] + bias, 0.0f);
          atomicAdd(node_in + (size_t)dn * NIN_W + colbase + j * 16 + ln, val);
        }
      }
    }
  }
}

// ---------------- node update MLP (WMMA) ----------------
// One wave handles 16 nodes: [16x640]@[640x64] -> [64x64] -> [64x128]
__global__ __launch_bounds__(EW * 32) void node_mlp_kernel(
    const float* __restrict__ node_in,
    const float* __restrict__ Wn1, const float* __restrict__ bn1,
    const float* __restrict__ Wn2, const float* __restrict__ bn2,
    const float* __restrict__ Wn3, const float* __restrict__ bn3,
    float* __restrict__ out) {
  __shared__ __align__(16) float Xs[EW][16 * XS_LD];
  __shared__ __align__(16) float Hs[EW][16 * HS_LD];

  const int tid  = threadIdx.x;
  const int w    = tid >> 5;
  const int lane = tid & 31;
  const int half = lane >> 4;
  const int ln   = lane & 15;

  const int  base   = (blockIdx.x * EW + w) * 16;
  const bool active = (base < N_NODES);
  const int  valid  = active ? imin(16, N_NODES - base) : 0;

  float* xs = &Xs[w][0];
  float* hs = &Hs[w][0];

  // ---- layer 1: K=640 streamed in 5 chunks of 128 ----
  v8f c1[4];
#pragma unroll
  for (int j = 0; j < 4; ++j)
#pragma unroll
    for (int i = 0; i < 8; ++i) c1[j][i] = 0.0f;

  for (int kc = 0; kc < 5; ++kc) {
    __syncthreads();          // WAR: previous chunk's A-frag reads done
    if (active) {
      int row = lane >> 1;
      int rr  = imin(row, valid - 1);
      stage_row64(node_in + (size_t)(base + rr) * NIN_W + kc * 128 + (lane & 1) * 64,
                  xs + row * XS_LD + (lane & 1) * 64);
    }
    async_fence();
    __syncthreads();
    for (int kk = 0; kk < 32; ++kk) {
      int krow = kk * 4 + half * 2;
      v2f a = *(const v2f*)(xs + ln * XS_LD + krow);
      int gk = kc * 128 + krow;
#pragma unroll
      for (int j = 0; j < 4; ++j) {
        v2f b;
        b.x = Wn1[(size_t)(gk + 0) * H_DIM + j * 16 + ln];
        b.y = Wn1[(size_t)(gk + 1) * H_DIM + j * 16 + ln];
        c1[j] = wmma4(a, b, c1[j]);
      }
    }
  }
  {
#pragma unroll
    for (int j = 0; j < 4; ++j) {
      float bias = bn1[j * 16 + ln];
#pragma unroll
      for (int v = 0; v < 8; ++v)
        hs[(v + 8 * half) * HS_LD + j * 16 + ln] = fmaxf(c1[j][v] + bias, 0.0f);
    }
  }
  __syncthreads();

  // ---- layer 2 ----
  v8f c2[4];
#pragma unroll
  for (int j = 0; j < 4; ++j)
#pragma unroll
    for (int i = 0; i < 8; ++i) c2[j][i] = 0.0f;

  for (int kk = 0; kk < 16; ++kk) {
    int krow = kk * 4 + half * 2;
    v2f a = *(const v2f*)(hs + ln * HS_LD + krow);
#pragma unroll
    for (int j = 0; j < 4; ++j) {
      v2f b;
      b.x = Wn2[(size_t)(krow + 0) * H_DIM + j * 16 + ln];
      b.y = Wn2[(size_t)(krow + 1) * H_DIM + j * 16 + ln];
      c2[j] = wmma4(a, b, c2[j]);
    }
  }
  __syncthreads();
  {
#pragma unroll
    for (int j = 0; j < 4; ++j) {
      float bias = bn2[j * 16 + ln];
#pragma unroll
      for (int v = 0; v < 8; ++v)
        xs[(v + 8 * half) * XS_LD + j * 16 + ln] = fmaxf(c2[j][v] + bias, 0.0f);
    }
  }
  __syncthreads();

  // ---- layer 3 (linear) ----
  v8f c3[8];
#pragma unroll
  for (int j = 0; j < 8; ++j)
#pragma unroll
    for (int i = 0; i < 8; ++i) c3[j][i] = 0.0f;

  for (int kk = 0; kk < 16; ++kk) {
    int krow = kk * 4 + half * 2;
    v2f a = *(const v2f*)(xs + ln * XS_LD + krow);
#pragma unroll
    for (int j = 0; j < 8; ++j) {
      v2f b;
      b.x = Wn3[(size_t)(krow + 0) * D_DIM + j * 16 + ln];
      b.y = Wn3[(size_t)(krow + 1) * D_DIM + j * 16 + ln];
      c3[j] = wmma4(a, b, c3[j]);
    }
  }

  if (active) {
#pragma unroll
    for (int j = 0; j < 8; ++j) {
      float bias = bn3[j * 16 + ln];
#pragma unroll
      for (int v = 0; v < 8; ++v) {
        int m = v + 8 * half;
        if (m < valid)
          out[(size_t)(base + m) * D_DIM + j * 16 + ln] = c3[j][v] + bias;
      }
    }
  }
}

// ---------------- host launcher ----------------
extern "C" void kernel_launch(void* const* d_in, const int* in_sizes, int n_in,
                              void* d_out, int out_size, void* d_ws, size_t ws_size,
                              hipStream_t stream) {
  const float* nodef = (const float*)d_in[0];
  const int*   esrc  = (const int*)d_in[1];
  const int*   edst  = (const int*)d_in[2];
  const int*   etype = (const int*)d_in[3];
  const float* Wr1   = (const float*)d_in[4];
  const float* br1   = (const float*)d_in[5];
  const float* Wr2   = (const float*)d_in[6];
  const float* br2   = (const float*)d_in[7];
  const float* Wr3   = (const float*)d_in[8];
  const float* br3   = (const float*)d_in[9];
  const float* Wn1   = (const float*)d_in[10];
  const float* bn1   = (const float*)d_in[11];
  const float* Wn2   = (const float*)d_in[12];
  const float* bn2   = (const float*)d_in[13];
  const float* Wn3   = (const float*)d_in[14];
  const float* bn3   = (const float*)d_in[15];
  float* out = (float*)d_out;

  // workspace layout: node_in [N,640] f32 | bucket [E] i32 | meta [16] i32
  char*  ws       = (char*)d_ws;
  float* node_in  = (float*)ws;
  size_t nin_b    = (size_t)N_NODES * NIN_W * sizeof(float);
  int*   bucket   = (int*)(ws + nin_b);
  int*   meta     = bucket + N_EDGES;

  zero_meta_kernel<<<1, 32, 0, stream>>>(meta);
  init_nodein_kernel<<<N_NODES, NIN_W, 0, stream>>>(nodef, node_in);
  count_kernel<<<(N_EDGES + 255) / 256, 256, 0, stream>>>(etype, meta);
  scan_kernel<<<1, 1, 0, stream>>>(meta);
  scatter_kernel<<<(N_EDGES + 255) / 256, 256, 0, stream>>>(etype, meta, bucket);
  {
    int max_groups = (N_EDGES + 15) / 16 + R_REL;  // upper bound on sum of per-relation groups
    int blocks = (max_groups + EW - 1) / EW;
    edge_mlp_kernel<<<blocks, EW * 32, 0, stream>>>(nodef, esrc, edst,
                                                    Wr1, br1, Wr2, br2, Wr3, br3,
                                                    bucket, meta, node_in);
  }
  {
    int blocks = (N_NODES + EW * 16 - 1) / (EW * 16);
    node_mlp_kernel<<<blocks, EW * 32, 0, stream>>>(node_in, Wn1, bn1, Wn2, bn2,
                                                    Wn3, bn3, out);
  }
}